// COLoss_45457933860953
// MI455X (gfx1250) — compile-verified
//
#include <hip/hip_runtime.h>
#include <hip/hip_bf16.h>

// Problem constants (match reference: B=16, C=2, H=W=512)
#define NB        16
#define HW        (512 * 512)            // 262144 pixels per batch image
#define BPB       64                     // blocks per batch
#define THREADS   256                    // 8 waves per block (wave32)
#define NBLOCKS   (NB * BPB)             // 1024 blocks
#define F4_PER_BLOCK (HW / BPB / 4)      // 1024 float4 groups per block
#define ITERS     (F4_PER_BLOCK / THREADS) // 4 iterations
#define BGW       0.4f

typedef float v2f __attribute__((ext_vector_type(2)));
typedef float v8f __attribute__((ext_vector_type(8)));

// Sum all 32 lane values of a wave with one V_WMMA_F32_16X16X4_F32.
// A (16x4): VGPR0 = lane value (K=0 for lanes 0-15, K=2 for lanes 16-31),
// VGPR1 = 0. B (4x16) = all ones  =>  D[m][n] = acc[m] + acc[m+16].
// Lane L<16 holds D rows 0..7 in its 8 D VGPRs, lane L>=16 rows 8..15,
// so (sum of 8 D regs) + xor-16 shuffle = full 32-lane sum, in every lane.
__device__ __forceinline__ float wave_sum_wmma(float v) {
    v2f a; a[0] = v;    a[1] = 0.0f;
    v2f b; b[0] = 1.0f; b[1] = 1.0f;
    v8f c = {};
    v8f d = __builtin_amdgcn_wmma_f32_16x16x4_f32(
        /*neg_a=*/false, a, /*neg_b=*/false, b,
        /*c_mod=*/(short)0, c, /*reuse_a=*/false, /*reuse_b=*/false);
    float s = ((d[0] + d[1]) + (d[2] + d[3])) + ((d[4] + d[5]) + (d[6] + d[7]));
    s += __shfl_xor(s, 16, 32);
    return s;
}

__global__ __launch_bounds__(THREADS)
void coloss_partials(const float* __restrict__ conf,
                     const float* __restrict__ off,
                     const int*   __restrict__ inst,
                     const float* __restrict__ gto,
                     float*       __restrict__ wsf) {
    const int bid = blockIdx.x;
    const int b   = bid / BPB;          // batch index
    const int s   = bid % BPB;          // slice within batch

    const size_t base1 = (size_t)b * HW;        // 1-channel plane base
    const size_t base2 = (size_t)b * 2 * HW;    // 2-channel tensor base

    const float4* __restrict__ c0 = (const float4*)(conf + base2);
    const float4* __restrict__ c1 = (const float4*)(conf + base2 + HW);
    const float4* __restrict__ o0 = (const float4*)(off  + base2);
    const float4* __restrict__ o1 = (const float4*)(off  + base2 + HW);
    const float4* __restrict__ g0 = (const float4*)(gto  + base2);
    const float4* __restrict__ g1 = (const float4*)(gto  + base2 + HW);
    const int4*   __restrict__ iv = (const int4*)(inst + base1);

    const int f4base = s * F4_PER_BLOCK + threadIdx.x;

    float ent = 0.0f;   // entropy accumulator
    float sse = 0.0f;   // fg squared-error accumulator
    float cnt = 0.0f;   // fg pixel count

#pragma unroll
    for (int it = 0; it < ITERS; ++it) {
        const int idx = f4base + it * THREADS;   // float4-granular index
        const int4   I  = iv[idx];
        const float4 C0 = c0[idx], C1 = c1[idx];
        const float4 O0 = o0[idx], O1 = o1[idx];
        const float4 G0 = g0[idx], G1 = g1[idx];

#define PIX(comp)                                                        \
        {                                                                \
            const int   ii = I.comp;                                     \
            const float cc = (ii != 0) ? C1.comp : C0.comp;              \
            const float e  = -__logf(cc);                                \
            ent += (ii == 0) ? e * BGW : e;                              \
            const float m  = (ii != 0) ? 1.0f : 0.0f;                    \
            const float dx = G0.comp - O0.comp;                          \
            const float dy = G1.comp - O1.comp;                          \
            sse += m * (dx * dx + dy * dy);                              \
            cnt += m;                                                    \
        }
        PIX(x) PIX(y) PIX(z) PIX(w)
#undef PIX
    }

    // Block reduction: WMMA per-wave sum, then 8 wave partials via LDS.
    // All 256 threads reach here (EXEC all ones, required for WMMA).
    const float we = wave_sum_wmma(ent);
    const float wo = wave_sum_wmma(sse);
    const float wc = wave_sum_wmma(cnt);

    __shared__ float red[3][THREADS / 32];
    const int wave = threadIdx.x >> 5;
    const int lane = threadIdx.x & 31;
    if (lane == 0) { red[0][wave] = we; red[1][wave] = wo; red[2][wave] = wc; }
    __syncthreads();
    if (threadIdx.x == 0) {
        float e = 0.0f, o = 0.0f, c = 0.0f;
#pragma unroll
        for (int w = 0; w < THREADS / 32; ++w) {
            e += red[0][w]; o += red[1][w]; c += red[2][w];
        }
        wsf[bid]               = e;   // entropy partial
        wsf[NBLOCKS + bid]     = o;   // offset-SSE partial (this batch only)
        wsf[2 * NBLOCKS + bid] = c;   // fg count partial
    }
}

__global__ void coloss_finalize(const float* __restrict__ wsf,
                                float* __restrict__ out) {
    __shared__ double sh[NB + 1];
    const int t = threadIdx.x;
    if (t < NB) {
        // per-batch: sum this batch's 64 block partials in double, fixed order
        double so = 0.0, sc = 0.0;
        for (int k = 0; k < BPB; ++k) {
            so += (double)wsf[NBLOCKS + t * BPB + k];
            sc += (double)wsf[2 * NBLOCKS + t * BPB + k];
        }
        sh[t] = (sc > 0.0) ? (so / sc) : 0.0;
    } else if (t == NB) {
        double se = 0.0;
        for (int k = 0; k < NBLOCKS; ++k) se += (double)wsf[k];
        sh[NB] = se;
    }
    __syncthreads();
    if (t == 0) {
        double offl = 0.0;
        for (int b = 0; b < NB; ++b) offl += sh[b];
        const double conf_loss = sh[NB] / ((double)NB * (double)HW);
        out[0] = (float)(conf_loss + offl / (double)NB);
    }
}

extern "C" void kernel_launch(void* const* d_in, const int* in_sizes, int n_in,
                              void* d_out, int out_size, void* d_ws, size_t ws_size,
                              hipStream_t stream) {
    const float* conf = (const float*)d_in[0];   // [16,2,512,512] f32
    const float* off  = (const float*)d_in[1];   // [16,2,512,512] f32
    const int*   inst = (const int*)  d_in[2];   // [16,1,512,512] i32
    const float* gto  = (const float*)d_in[3];   // [16,2,512,512] f32
    float* out = (float*)d_out;
    float* wsf = (float*)d_ws;                   // 3 * 1024 floats = 12 KB

    coloss_partials<<<NBLOCKS, THREADS, 0, stream>>>(conf, off, inst, gto, wsf);
    coloss_finalize<<<1, 64, 0, stream>>>(wsf, out);
}